// DKT_GAT_53927609368623
// MI455X (gfx1250) — compile-verified
//
#include <hip/hip_runtime.h>
#include <math.h>

typedef float v2f __attribute__((ext_vector_type(2)));
typedef float v8f __attribute__((ext_vector_type(8)));

#define N_SK   10000
#define EMBD   128
#define HEADS  4
#define DHD    32
#define BB     256
#define TT     200
#define RNNH   256
#define G3H    768
#define NEG_SLOPE 0.2f

// ---------- WMMA helper: D = A(16x4 f32) * B(4x16 f32) + C, exact fp32 ----------
__device__ __forceinline__ v8f wmma4(v2f a, v2f b, v8f c) {
  return __builtin_amdgcn_wmma_f32_16x16x4_f32(false, a, false, b, (short)0, c,
                                               false, false);
}

// ---------- float atomic max via int/uint trick ----------
__device__ __forceinline__ void atomicMaxF(float* addr, float val) {
  if (val >= 0.0f) atomicMax((int*)addr, __float_as_int(val));
  else             atomicMin((unsigned int*)addr, __float_as_uint(val));
}

// ---------- generic fill ----------
__global__ void k_fill(float* __restrict__ p, float v, long n) {
  long i = (long)blockIdx.x * blockDim.x + threadIdx.x;
  for (; i < n; i += (long)gridDim.x * blockDim.x) p[i] = v;
}

// ---------- self loop stats ----------
__global__ void k_loop_stats(const int* __restrict__ ei,
                             const float* __restrict__ ew,
                             float* __restrict__ s, float* __restrict__ c, int E) {
  int e = blockIdx.x * blockDim.x + threadIdx.x;
  if (e >= E) return;
  int dst = ei[E + e];
  atomicAdd(&s[dst], ew[e]);
  atomicAdd(&c[dst], 1.0f);
}

__global__ void k_loop_attr(const float* __restrict__ s,
                            const float* __restrict__ c,
                            float* __restrict__ la, int n) {
  int i = blockIdx.x * blockDim.x + threadIdx.x;
  if (i >= n) return;
  la[i] = s[i] / fmaxf(c[i], 1.0f);
}

// ---------- WMMA GEMM: C[M,N] = A[M,K] @ W[N,K]^T + bias[N] ----------
// one wave per 16x16 tile; grid.x = M/16, grid.y = N/16
__global__ void k_gemm_nt_bias(const float* __restrict__ A,
                               const float* __restrict__ W,
                               const float* __restrict__ bias,
                               float* __restrict__ C,
                               int K, int lda, int ldw, int ldc) {
  int m0 = blockIdx.x * 16;
  int n0 = blockIdx.y * 16;
  int l = threadIdx.x;
  int half = l >> 4, lm = l & 15;
  v8f acc = {};
  const float* ap = A + (long)(m0 + lm) * lda + half * 2;
  const float* wp = W + (long)(n0 + lm) * ldw + half * 2;
  for (int k = 0; k < K; k += 4) {
    v2f a = *(const v2f*)(ap + k);
    v2f b = *(const v2f*)(wp + k);
    acc = wmma4(a, b, acc);
  }
  float bi = bias ? bias[n0 + lm] : 0.0f;
  for (int v = 0; v < 8; ++v) {
    int m = m0 + half * 8 + v;
    C[(long)m * ldc + n0 + lm] = acc[v] + bi;
  }
}

// ---------- GATv2 per-edge attention logits + segment max ----------
__global__ void k_edge_logits(const int* __restrict__ ei, const float* __restrict__ ew,
                              const float* __restrict__ loop_attr,
                              const float* __restrict__ xl, const float* __restrict__ xr,
                              const float* __restrict__ We, const float* __restrict__ att,
                              float* __restrict__ elog, float* __restrict__ emax,
                              int E, int E2) {
  int e = blockIdx.x * blockDim.x + threadIdx.x;
  if (e >= E2) return;
  int src, dst; float w;
  if (e < E) { src = ei[e]; dst = ei[E + e]; w = ew[e]; }
  else       { src = dst = e - E; w = loop_attr[src]; }
  const float* xls = xl + (long)src * EMBD;
  const float* xrd = xr + (long)dst * EMBD;
  for (int h = 0; h < HEADS; ++h) {
    float acc = 0.0f;
    for (int d = 0; d < DHD; ++d) {
      int k = h * DHD + d;
      float m = xls[k] + xrd[k] + w * We[k];
      float ma = m > 0.0f ? m : NEG_SLOPE * m;
      acc += ma * att[k];
    }
    elog[(long)e * HEADS + h] = acc;
    atomicMaxF(&emax[(long)dst * HEADS + h], acc);
  }
}

// ---------- exp(e - max) and segment-sum denom ----------
__global__ void k_edge_exp(const int* __restrict__ ei,
                           float* __restrict__ elog,
                           const float* __restrict__ emax,
                           float* __restrict__ denom, int E, int E2) {
  int e = blockIdx.x * blockDim.x + threadIdx.x;
  if (e >= E2) return;
  int dst = (e < E) ? ei[E + e] : (e - E);
  for (int h = 0; h < HEADS; ++h) {
    float a = expf(elog[(long)e * HEADS + h] - emax[(long)dst * HEADS + h]);
    elog[(long)e * HEADS + h] = a;
    atomicAdd(&denom[(long)dst * HEADS + h], a);
  }
}

// ---------- scatter alpha * xl[src] into acc[dst] ----------
__global__ void k_edge_scatter(const int* __restrict__ ei,
                               const float* __restrict__ elog,
                               const float* __restrict__ denom,
                               const float* __restrict__ xl,
                               float* __restrict__ accb, int E, int E2) {
  int gid = blockIdx.x * blockDim.x + threadIdx.x;
  if (gid >= E2 * HEADS) return;
  int e = gid >> 2, h = gid & 3;
  int src, dst;
  if (e < E) { src = ei[e]; dst = ei[E + e]; }
  else       { src = dst = e - E; }
  float alpha = elog[(long)e * HEADS + h] /
                (denom[(long)dst * HEADS + h] + 1e-16f);
  const float* xs = xl + (long)src * EMBD + h * DHD;
  float* ao = accb + (long)dst * EMBD + h * DHD;
  for (int d = 0; d < DHD; ++d) atomicAdd(&ao[d], alpha * xs[d]);
}

// ---------- out = elu(acc + bias) ----------
__global__ void k_finalize_elu(const float* __restrict__ accb,
                               const float* __restrict__ bias,
                               float* __restrict__ out, long n) {
  long i = (long)blockIdx.x * blockDim.x + threadIdx.x;
  if (i >= n) return;
  float v = accb[i] + bias[i & (EMBD - 1)];
  out[i] = v > 0.0f ? v : (expf(v) - 1.0f);
}

// ---------- gather GRU inputs ----------
__global__ void k_gather_x(const int* __restrict__ skills, const int* __restrict__ resp,
                           const float* __restrict__ skill_embs,
                           const float* __restrict__ resp_emb,
                           float* __restrict__ x, long n) {
  long i = (long)blockIdx.x * blockDim.x + threadIdx.x;
  if (i >= n) return;
  long bt = i >> 7;
  int k = (int)(i & 127);
  x[i] = skill_embs[(long)skills[bt] * EMBD + k] + resp_emb[(long)resp[bt] * EMBD + k];
}

// ---------- transpose: in[R,C] -> out[C,R] ----------
__global__ void k_transpose(const float* __restrict__ in, float* __restrict__ out,
                            int R, int C) {
  long i = (long)blockIdx.x * blockDim.x + threadIdx.x;
  if (i >= (long)R * C) return;
  int r = (int)(i / C), c = (int)(i % C);
  out[(long)c * R + r] = in[i];
}

// ---------- fused GRU + projection, 16 batch rows per block ----------
#define HS 260   // padded stride for 256-wide LDS rows (bank spread, 8B aligned)
#define XS 132   // padded stride for 128-wide rows
#define GS 772   // padded stride for 768-wide rows

__global__ void __launch_bounds__(256)
k_gru_proj(const float* __restrict__ x, const int* __restrict__ lengths,
           const float* __restrict__ WihT, const float* __restrict__ WhhT,
           const float* __restrict__ bih, const float* __restrict__ bhh,
           const float* __restrict__ projWT, const float* __restrict__ projB,
           float* __restrict__ out) {
  extern __shared__ float sm[];
  float* h_cur = sm;                 // [16][HS]
  float* g     = h_cur + 16 * HS;    // [16][GS]  xg+bih (+hg for r,z)
  float* hnb   = g + 16 * GS;        // [16][HS]  hg for n-gate
  float* xt    = hnb + 16 * HS;      // [16][XS]
  float* yb    = xt + 16 * XS;       // [16][HS]

  int tid = threadIdx.x;
  int wave = tid >> 5, lane = tid & 31;
  int half = lane >> 4, lm = lane & 15;
  int b0 = blockIdx.x * 16;

  for (int i = tid; i < 16 * HS; i += 256) h_cur[i] = 0.0f;
  __syncthreads();

  for (int t = 0; t < TT; ++t) {
    // load x_t tile
    for (int i = tid; i < 16 * EMBD; i += 256) {
      int r = i >> 7, c = i & 127;
      xt[r * XS + c] = x[((long)(b0 + r) * TT + t) * EMBD + c];
    }
    __syncthreads();

    // GEMM1: g = x_t @ WihT + bih   (48 N-tiles / 8 waves)
    for (int it = 0; it < 6; ++it) {
      int n = (wave + it * 8) * 16 + lm;
      v8f acc = {};
      const float* ap = xt + lm * XS + half * 2;
      for (int k = 0; k < EMBD; k += 4) {
        v2f a = *(const v2f*)(ap + k);
        int kk = k + half * 2;
        v2f b = { WihT[(long)kk * G3H + n], WihT[(long)(kk + 1) * G3H + n] };
        acc = wmma4(a, b, acc);
      }
      float bi = bih[n];
      for (int v = 0; v < 8; ++v) g[(half * 8 + v) * GS + n] = acc[v] + bi;
    }
    __syncthreads();

    // GEMM2: hg = h @ WhhT + bhh; fuse r,z into g; keep n-part separate
    for (int it = 0; it < 6; ++it) {
      int n = (wave + it * 8) * 16 + lm;
      v8f acc = {};
      const float* ap = h_cur + lm * HS + half * 2;
      for (int k = 0; k < RNNH; k += 4) {
        v2f a = *(const v2f*)(ap + k);
        int kk = k + half * 2;
        v2f b = { WhhT[(long)kk * G3H + n], WhhT[(long)(kk + 1) * G3H + n] };
        acc = wmma4(a, b, acc);
      }
      float bi = bhh[n];
      if (n < 2 * RNNH) {
        for (int v = 0; v < 8; ++v) g[(half * 8 + v) * GS + n] += acc[v] + bi;
      } else {
        for (int v = 0; v < 8; ++v) hnb[(half * 8 + v) * HS + (n - 2 * RNNH)] = acc[v] + bi;
      }
    }
    __syncthreads();

    // gates + masking
    for (int i = tid; i < 16 * RNNH; i += 256) {
      int r = i >> 8, c = i & 255;
      float xr_ = g[r * GS + c];
      float xz_ = g[r * GS + RNNH + c];
      float xn_ = g[r * GS + 2 * RNNH + c];
      float hn_ = hnb[r * HS + c];
      float rg = 1.0f / (1.0f + expf(-xr_));
      float zg = 1.0f / (1.0f + expf(-xz_));
      float ng = tanhf(xn_ + rg * hn_);
      float hp = h_cur[r * HS + c];
      float hnew = (1.0f - zg) * ng + zg * hp;
      bool msk = t < lengths[b0 + r];
      h_cur[r * HS + c] = msk ? hnew : hp;
      yb[r * HS + c] = msk ? hnew : 0.0f;
    }
    __syncthreads();

    // projection: out_t[16,128] = y @ projWT + projB  (8 tiles / 8 waves)
    {
      int n = wave * 16 + lm;
      v8f acc = {};
      const float* ap = yb + lm * HS + half * 2;
      for (int k = 0; k < RNNH; k += 4) {
        v2f a = *(const v2f*)(ap + k);
        int kk = k + half * 2;
        v2f b = { projWT[(long)kk * EMBD + n], projWT[(long)(kk + 1) * EMBD + n] };
        acc = wmma4(a, b, acc);
      }
      float bi = projB[n];
      for (int v = 0; v < 8; ++v) {
        int m = half * 8 + v;
        out[((long)(b0 + m) * TT + t) * EMBD + n] = acc[v] + bi;
      }
    }
    __syncthreads();
  }
}

// =====================================================================

static void run_gat_layer(const float* h_in, float* h_out,
                          const int* ei, const float* ew, const float* loop_attr,
                          const float* Wl, const float* bl,
                          const float* Wr, const float* br,
                          const float* We, const float* att, const float* bias,
                          float* xl, float* xr, float* emax, float* denom,
                          float* accb, float* elog,
                          int E, int E2, hipStream_t stream) {
  dim3 gg(N_SK / 16, EMBD / 16);
  k_gemm_nt_bias<<<gg, 32, 0, stream>>>(h_in, Wl, bl, xl, EMBD, EMBD, EMBD, EMBD);
  k_gemm_nt_bias<<<gg, 32, 0, stream>>>(h_in, Wr, br, xr, EMBD, EMBD, EMBD, EMBD);
  k_fill<<<256, 256, 0, stream>>>(emax, -3.0e38f, (long)N_SK * HEADS);
  k_fill<<<256, 256, 0, stream>>>(denom, 0.0f, (long)N_SK * HEADS);
  k_fill<<<1024, 256, 0, stream>>>(accb, 0.0f, (long)N_SK * EMBD);
  int gE = (E2 + 255) / 256;
  k_edge_logits<<<gE, 256, 0, stream>>>(ei, ew, loop_attr, xl, xr, We, att,
                                        elog, emax, E, E2);
  k_edge_exp<<<gE, 256, 0, stream>>>(ei, elog, emax, denom, E, E2);
  k_edge_scatter<<<(E2 * HEADS + 255) / 256, 256, 0, stream>>>(ei, elog, denom,
                                                               xl, accb, E, E2);
  long nf = (long)N_SK * EMBD;
  k_finalize_elu<<<(unsigned)((nf + 255) / 256), 256, 0, stream>>>(accb, bias, h_out, nf);
}

extern "C" void kernel_launch(void* const* d_in, const int* in_sizes, int n_in,
                              void* d_out, int out_size, void* d_ws, size_t ws_size,
                              hipStream_t stream) {
  const int*   skills_in = (const int*)d_in[0];
  const int*   resp_in   = (const int*)d_in[1];
  const int*   lengths   = (const int*)d_in[2];
  const int*   edge_idx  = (const int*)d_in[3];
  const float* edge_w    = (const float*)d_in[4];
  const float* emb       = (const float*)d_in[5];
  const float* g0_Wl = (const float*)d_in[6],  *g0_bl = (const float*)d_in[7];
  const float* g0_Wr = (const float*)d_in[8],  *g0_br = (const float*)d_in[9];
  const float* g0_We = (const float*)d_in[10], *g0_att = (const float*)d_in[11];
  const float* g0_bias = (const float*)d_in[12];
  const float* g1_Wl = (const float*)d_in[13], *g1_bl = (const float*)d_in[14];
  const float* g1_Wr = (const float*)d_in[15], *g1_br = (const float*)d_in[16];
  const float* g1_We = (const float*)d_in[17], *g1_att = (const float*)d_in[18];
  const float* g1_bias = (const float*)d_in[19];
  const float* resp_emb = (const float*)d_in[20];
  const float* Wih = (const float*)d_in[21], *Whh = (const float*)d_in[22];
  const float* bih = (const float*)d_in[23], *bhh = (const float*)d_in[24];
  const float* projW = (const float*)d_in[25], *projB = (const float*)d_in[26];

  const int E  = in_sizes[3] / 2;
  const int E2 = E + N_SK;

  float* out       = (float*)d_out;
  float* skill_out = out + (long)BB * TT * EMBD;   // second output region

  // workspace carve-up
  float* p = (float*)d_ws;
  float* loop_s    = p; p += N_SK;
  float* loop_c    = p; p += N_SK;
  float* loop_attr = p; p += N_SK;
  float* xl   = p; p += (long)N_SK * EMBD;
  float* xr   = p; p += (long)N_SK * EMBD;
  float* h1   = p; p += (long)N_SK * EMBD;
  float* emax = p; p += (long)N_SK * HEADS;
  float* denom= p; p += (long)N_SK * HEADS;
  float* accb = p; p += (long)N_SK * EMBD;
  float* elog = p; p += (long)E2 * HEADS;
  float* xseq = p; p += (long)BB * TT * EMBD;
  float* WihT = p; p += (long)EMBD * G3H;
  float* WhhT = p; p += (long)RNNH * G3H;
  float* projWT = p; p += (long)RNNH * EMBD;

  // self-loop mean edge weights
  k_fill<<<64, 256, 0, stream>>>(loop_s, 0.0f, 2L * N_SK);
  k_loop_stats<<<(E + 255) / 256, 256, 0, stream>>>(edge_idx, edge_w, loop_s, loop_c, E);
  k_loop_attr<<<(N_SK + 255) / 256, 256, 0, stream>>>(loop_s, loop_c, loop_attr, N_SK);

  // GAT layer 0: emb -> h1 ; layer 1: h1 -> skill_out (second output region)
  run_gat_layer(emb, h1, edge_idx, edge_w, loop_attr,
                g0_Wl, g0_bl, g0_Wr, g0_br, g0_We, g0_att, g0_bias,
                xl, xr, emax, denom, accb, elog, E, E2, stream);
  run_gat_layer(h1, skill_out, edge_idx, edge_w, loop_attr,
                g1_Wl, g1_bl, g1_Wr, g1_br, g1_We, g1_att, g1_bias,
                xl, xr, emax, denom, accb, elog, E, E2, stream);

  // GRU input sequence
  long nx = (long)BB * TT * EMBD;
  k_gather_x<<<(unsigned)((nx + 255) / 256), 256, 0, stream>>>(
      skills_in, resp_in, skill_out, resp_emb, xseq, nx);

  // pre-transpose weights for coalesced WMMA B loads
  k_transpose<<<((G3H * EMBD) + 255) / 256, 256, 0, stream>>>(Wih, WihT, G3H, EMBD);
  k_transpose<<<((G3H * RNNH) + 255) / 256, 256, 0, stream>>>(Whh, WhhT, G3H, RNNH);
  k_transpose<<<((EMBD * RNNH) + 255) / 256, 256, 0, stream>>>(projW, projWT, EMBD, RNNH);

  // fused GRU + projection
  size_t smem = (size_t)(16 * HS * 3 + 16 * GS + 16 * XS) * sizeof(float);
  k_gru_proj<<<BB / 16, 256, smem, stream>>>(xseq, lengths, WihT, WhhT,
                                             bih, bhh, projWT, projB, out);
}